// EPIC_78228534329710
// MI455X (gfx1250) — compile-verified
//
#include <hip/hip_runtime.h>
#include <math.h>

#define NGN 20000
#define NPT 8000
#define HD  64
#define EPPI 300000
#define EMUT 600000
#define NEG_SLOPE 0.2f

typedef float v2f __attribute__((ext_vector_type(2)));
typedef float v8f __attribute__((ext_vector_type(8)));

// ---------------------------------------------------------------------------
// Y[N,64] = act(X[idx?][ ,64] @ W[64,64] + b)   via V_WMMA_F32_16X16X4_F32
// block = 256 threads = 8 waves; each wave -> one 16-row x 64-col strip.
// ---------------------------------------------------------------------------
__global__ void k_gemm64_wmma(const float* __restrict__ X, const int* __restrict__ idx,
                              const float* __restrict__ W, const float* __restrict__ bias,
                              float* __restrict__ Y, int N, int doRelu)
{
    __shared__ float Ws[64 * 64];
    int t = threadIdx.x;
    {   // stage W (16 KB) into LDS: 256 threads x 4 float4
        const float4* Wv = (const float4*)W;
        float4* Sv = (float4*)Ws;
#pragma unroll
        for (int i = 0; i < 4; ++i) Sv[t + 256 * i] = Wv[t + 256 * i];
    }
    __syncthreads();

    int wave = t >> 5;
    int lane = t & 31;
    int half = lane >> 4;      // 0: K pair (k0,k0+1)   1: (k0+2,k0+3)
    int r    = lane & 15;
    int tile = blockIdx.x * 8 + wave;
    int row0 = tile * 16;
    if (row0 >= N) return;     // wave-uniform exit (EXEC stays all-1 for WMMA)

    int arow = row0 + r;
    if (arow >= N) arow = N - 1;
    long xoff = (long)(idx ? idx[arow] : arow) * HD;
    const v2f* Xrow = (const v2f*)(X + xoff);

    v8f acc[4] = {};           // 4 col tiles of 16, C = 16x16 f32 = 8 VGPRs each
#pragma unroll
    for (int k0 = 0; k0 < 64; k0 += 4) {
        // A fragment: lanes0-15 -> A[r][k0],A[r][k0+1]; lanes16-31 -> A[r][k0+2],A[r][k0+3]
        v2f a = Xrow[(k0 >> 1) + half];
#pragma unroll
        for (int nt = 0; nt < 4; ++nt) {
            int col = nt * 16 + r;
            v2f b;
            b.x = Ws[(k0 + 2 * half) * 64 + col];
            b.y = Ws[(k0 + 2 * half + 1) * 64 + col];
            acc[nt] = __builtin_amdgcn_wmma_f32_16x16x4_f32(
                false, a, false, b, (short)0, acc[nt], false, false);
        }
    }

    // C layout: VGPR v -> (M=v, N=lane) lanes0-15 ; (M=8+v, N=lane-16) lanes16-31
#pragma unroll
    for (int nt = 0; nt < 4; ++nt) {
        int col = nt * 16 + r;
        float bb = bias[col];
#pragma unroll
        for (int v = 0; v < 8; ++v) {
            int orow = row0 + v + 8 * half;
            float val = acc[nt][v] + bb;
            if (doRelu) val = fmaxf(val, 0.0f);
            if (orow < N) Y[(long)orow * HD + col] = val;
        }
    }
}

// u[j] = sum_h rowvec[h]*We[h][j] ; v[j] = sum_h be[h]*We[h][j]   (rank-1 edge encoder)
__global__ void k_uv(const float* __restrict__ rowvec, const float* __restrict__ be,
                     const float* __restrict__ We, float* __restrict__ u, float* __restrict__ v)
{
    int j = threadIdx.x;  // 64 threads
    float su = 0.f, sv = 0.f;
#pragma unroll 8
    for (int h = 0; h < 64; ++h) {
        float w = We[h * 64 + j];
        su = fmaf(rowvec[h], w, su);
        sv = fmaf(be[h], w, sv);
    }
    u[j] = su; v[j] = sv;
}

__global__ void k_fill(float* __restrict__ p, float val, int n)
{
    int i = blockIdx.x * blockDim.x + threadIdx.x;
    if (i < n) p[i] = val;
}

// out[i] = b1[i&63] (+ b2[i&63])   -- conv bias init of flow accumulators
__global__ void k_init_bias2(float* __restrict__ out, const float* __restrict__ b1,
                             const float* __restrict__ b2, int total)
{
    int i = blockIdx.x * blockDim.x + threadIdx.x;
    if (i >= total) return;
    int h = i & 63;
    float v = b1[h];
    if (b2) v += b2[h];
    out[i] = v;
}

__device__ __forceinline__ float lrelu(float m) { return m > 0.f ? m : NEG_SLOPE * m; }

// pass 1: logit[e] = att . leakyrelu(xl[src] + xr[dst] + attr*u + v); atomic float max into mx[dst]
__global__ void k_edge_logits(const int* __restrict__ src, const int* __restrict__ dst,
                              const float* __restrict__ attr,
                              const float* __restrict__ xl, const float* __restrict__ xr,
                              const float* __restrict__ u, const float* __restrict__ v,
                              const float* __restrict__ att,
                              float* __restrict__ elog, float* __restrict__ mx, int E)
{
    __shared__ float su[64], sv[64], sa[64];
    int t = threadIdx.x;
    if (t < 64) { su[t] = u[t]; sv[t] = v[t]; sa[t] = att[t]; }
    __syncthreads();
    int e = blockIdx.x * blockDim.x + t;
    if (e >= E) return;
    int s = src[e], d = dst[e];
    float a = attr[e];
    const float4* ps = (const float4*)(xl + (long)s * HD);
    const float4* pd = (const float4*)(xr + (long)d * HD);
    float acc = 0.f;
#pragma unroll 4
    for (int h4 = 0; h4 < 16; ++h4) {
        float4 A = ps[h4];
        float4 B = pd[h4];
        int h = h4 * 4;
        float m0 = lrelu(A.x + B.x + fmaf(a, su[h + 0], sv[h + 0]));
        float m1 = lrelu(A.y + B.y + fmaf(a, su[h + 1], sv[h + 1]));
        float m2 = lrelu(A.z + B.z + fmaf(a, su[h + 2], sv[h + 2]));
        float m3 = lrelu(A.w + B.w + fmaf(a, su[h + 3], sv[h + 3]));
        acc = fmaf(sa[h + 0], m0, acc);
        acc = fmaf(sa[h + 1], m1, acc);
        acc = fmaf(sa[h + 2], m2, acc);
        acc = fmaf(sa[h + 3], m3, acc);
    }
    elog[e] = acc;
    // float atomic max (ordered int/uint trick; mx initialized to -inf)
    if (acc >= 0.f) atomicMax((int*)(mx + d), __float_as_int(acc));
    else            atomicMin((unsigned int*)(mx + d), __float_as_uint(acc));
}

// pass 2: ex = exp(logit - mx[dst]); elog <- ex; den[dst] += ex
__global__ void k_edge_exp(const int* __restrict__ dst, float* __restrict__ elog,
                           const float* __restrict__ mx, float* __restrict__ den, int E)
{
    int e = blockIdx.x * blockDim.x + threadIdx.x;
    if (e >= E) return;
    int d = dst[e];
    float ex = __expf(elog[e] - mx[d]);
    elog[e] = ex;
    atomicAdd(den + d, ex);
}

// pass 3: alpha = ex/(den[dst]+1e-16); out[dst,:] += alpha * xl[src,:]
__global__ void k_edge_scatter(const int* __restrict__ src, const int* __restrict__ dst,
                               const float* __restrict__ elog, const float* __restrict__ den,
                               const float* __restrict__ xl, float* __restrict__ out, int E)
{
    int e = blockIdx.x * blockDim.x + threadIdx.x;
    if (e >= E) return;
    int s = src[e], d = dst[e];
    float alpha = elog[e] / (den[d] + 1e-16f);
    const float4* ps = (const float4*)(xl + (long)s * HD);
    float* po = out + (long)d * HD;
#pragma unroll 4
    for (int h4 = 0; h4 < 16; ++h4) {
        float4 A = ps[h4];
        atomicAdd(po + h4 * 4 + 0, alpha * A.x);
        atomicAdd(po + h4 * 4 + 1, alpha * A.y);
        atomicAdd(po + h4 * 4 + 2, alpha * A.z);
        atomicAdd(po + h4 * 4 + 3, alpha * A.w);
    }
}

// flow <- relu(flow) in place; node += flow
__global__ void k_relu_add(float* __restrict__ node, float* __restrict__ flow, int n)
{
    int i = blockIdx.x * blockDim.x + threadIdx.x;
    if (i >= n) return;
    float f = fmaxf(flow[i], 0.f);
    flow[i] = f;
    node[i] += f;
}

__global__ void k_copy4(float4* __restrict__ dst, const float4* __restrict__ src, int n4)
{
    int i = blockIdx.x * blockDim.x + threadIdx.x;
    if (i < n4) dst[i] = src[i];
}

// ---------------------------------------------------------------------------
extern "C" void kernel_launch(void* const* d_in, const int* in_sizes, int n_in,
                              void* d_out, int out_size, void* d_ws, size_t ws_size,
                              hipStream_t stream)
{
    (void)in_sizes; (void)n_in; (void)out_size; (void)ws_size;
    const int*   gene_x    = (const int*)d_in[0];
    const int*   patient_x = (const int*)d_in[1];
    const int*   ppi_src   = (const int*)d_in[2];
    const int*   ppi_dst   = (const int*)d_in[3];
    const int*   mut_pat   = (const int*)d_in[4];
    const int*   mut_gene  = (const int*)d_in[5];
    const float* ppi_attr  = (const float*)d_in[6];
    const float* mut_attr  = (const float*)d_in[7];
    // params: jax tree_flatten order (dict keys sorted; 'W*' < 'b*' < 'g*' < 'l*' < 'p*')
    const float* We_mut = (const float*)d_in[8];
    const float* We_ppi = (const float*)d_in[9];
    const float* Wg     = (const float*)d_in[10];
    const float* Wp     = (const float*)d_in[11];
    const float* be_mut = (const float*)d_in[12];
    const float* be_ppi = (const float*)d_in[13];
    const float* bg     = (const float*)d_in[14];
    const float* bp     = (const float*)d_in[15];
    const float* gene_emb = (const float*)d_in[16];
    struct Gat { const float *We, *Wl, *Wr, *att, *bias, *bl, *br; };
    Gat gat[2][3];  // [layer][0=mby,1=mut,2=ppi] -- dict keys sorted: mby<mut<ppi
    int pi = 17;
    for (int l = 0; l < 2; ++l)
        for (int c = 0; c < 3; ++c) {   // leaf keys sorted: We<Wl<Wr<att<bias<bl<br
            gat[l][c].We   = (const float*)d_in[pi++];
            gat[l][c].Wl   = (const float*)d_in[pi++];
            gat[l][c].Wr   = (const float*)d_in[pi++];
            gat[l][c].att  = (const float*)d_in[pi++];
            gat[l][c].bias = (const float*)d_in[pi++];
            gat[l][c].bl   = (const float*)d_in[pi++];
            gat[l][c].br   = (const float*)d_in[pi++];
        }
    const float* patient_emb = (const float*)d_in[pi];

    // workspace layout (f32): ~20 MB
    float* ws   = (float*)d_ws;
    float* g    = ws;                         // NGN*64
    float* pt   = g + (size_t)NGN * HD;       // NPT*64
    float* xl   = pt + (size_t)NPT * HD;      // NGN*64
    float* xr   = xl + (size_t)NGN * HD;      // NGN*64
    float* elog = xr + (size_t)NGN * HD;      // EMUT
    float* mx   = elog + EMUT;                // NGN
    float* den  = mx + NGN;                   // NGN
    float* uv   = den + NGN;                  // 2 layers * 3 convs * 128

    auto cdiv = [](int a, int b) { return (a + b - 1) / b; };
    const int TB = 256;

    // edge-encoder rank-1 vectors per conv: u at uv+idx*128, v at +64
    for (int l = 0; l < 2; ++l) {
        k_uv<<<1, 64, 0, stream>>>(We_ppi, be_ppi, gat[l][2].We, uv + (l * 3 + 2) * 128, uv + (l * 3 + 2) * 128 + 64);
        k_uv<<<1, 64, 0, stream>>>(We_mut, be_mut, gat[l][1].We, uv + (l * 3 + 1) * 128, uv + (l * 3 + 1) * 128 + 64);
        k_uv<<<1, 64, 0, stream>>>(We_mut, be_mut, gat[l][0].We, uv + (l * 3 + 0) * 128, uv + (l * 3 + 0) * 128 + 64);
    }

    // node transforms (WMMA)
    k_gemm64_wmma<<<cdiv(NGN, 128), TB, 0, stream>>>(gene_emb,    gene_x,    Wg, bg, g,  NGN, 1);
    k_gemm64_wmma<<<cdiv(NPT, 128), TB, 0, stream>>>(patient_emb, patient_x, Wp, bp, pt, NPT, 1);

    auto run_conv = [&](const float* xsrc, int Ns, const float* xdst, int Nd,
                        const int* esrc, const int* edst, const float* eattr, int E,
                        const Gat& P, const float* u, const float* v, float* outbuf) {
        k_gemm64_wmma<<<cdiv(Ns, 128), TB, 0, stream>>>(xsrc, nullptr, P.Wl, P.bl, xl, Ns, 0);
        k_gemm64_wmma<<<cdiv(Nd, 128), TB, 0, stream>>>(xdst, nullptr, P.Wr, P.br, xr, Nd, 0);
        k_fill<<<cdiv(Nd, TB), TB, 0, stream>>>(mx, -INFINITY, Nd);
        k_fill<<<cdiv(Nd, TB), TB, 0, stream>>>(den, 0.0f, Nd);
        k_edge_logits<<<cdiv(E, TB), TB, 0, stream>>>(esrc, edst, eattr, xl, xr, u, v, P.att, elog, mx, E);
        k_edge_exp<<<cdiv(E, TB), TB, 0, stream>>>(edst, elog, mx, den, E);
        k_edge_scatter<<<cdiv(E, TB), TB, 0, stream>>>(esrc, edst, elog, den, xl, outbuf, E);
    };

    for (int l = 0; l < 2; ++l) {
        float* flow_g = (float*)d_out + (size_t)(NGN + NPT) * HD * (1 + l);
        float* flow_p = flow_g + (size_t)NGN * HD;
        // init with conv biases (segment_sum + bias; ppi & mut share the gene accumulator)
        k_init_bias2<<<cdiv(NGN * HD, TB), TB, 0, stream>>>(flow_g, gat[l][2].bias, gat[l][1].bias, NGN * HD);
        k_init_bias2<<<cdiv(NPT * HD, TB), TB, 0, stream>>>(flow_p, gat[l][0].bias, nullptr, NPT * HD);
        // ppi: g -> g
        run_conv(g, NGN, g, NGN, ppi_src, ppi_dst, ppi_attr, EPPI,
                 gat[l][2], uv + (l * 3 + 2) * 128, uv + (l * 3 + 2) * 128 + 64, flow_g);
        // mut: pt -> g
        run_conv(pt, NPT, g, NGN, mut_pat, mut_gene, mut_attr, EMUT,
                 gat[l][1], uv + (l * 3 + 1) * 128, uv + (l * 3 + 1) * 128 + 64, flow_g);
        // mby: g -> pt
        run_conv(g, NGN, pt, NPT, mut_gene, mut_pat, mut_attr, EMUT,
                 gat[l][0], uv + (l * 3 + 0) * 128, uv + (l * 3 + 0) * 128 + 64, flow_p);
        k_relu_add<<<cdiv(NGN * HD, TB), TB, 0, stream>>>(g,  flow_g, NGN * HD);
        k_relu_add<<<cdiv(NPT * HD, TB), TB, 0, stream>>>(pt, flow_p, NPT * HD);
    }

    // final g, pt -> d_out head
    k_copy4<<<cdiv(NGN * HD / 4, TB), TB, 0, stream>>>((float4*)d_out, (const float4*)g, NGN * HD / 4);
    k_copy4<<<cdiv(NPT * HD / 4, TB), TB, 0, stream>>>((float4*)d_out + (size_t)NGN * 16, (const float4*)pt, NPT * HD / 4);
}